// MLPPredictor_58317065945293
// MI455X (gfx1250) — compile-verified
//
#include <hip/hip_runtime.h>

typedef __attribute__((ext_vector_type(2))) float v2f;
typedef __attribute__((ext_vector_type(8))) float v8f;

#define EDGES_PER_TILE 16
#define WAVES_PER_BLOCK 8
#define D 128            // feature dim per node (reference: 128)

// One wave32 per 16-edge tile.
// A (16x256 f32, gathered edge features) x B (256x2 f32 = W^T, zero-padded to
// 256x16) accumulated via 64 chained V_WMMA_F32_16X16X4_F32 ops, bias folded
// into C init, then stable 2-class log-softmax via lane-pair shuffle.
__global__ __launch_bounds__(256, 1) void mlp_edge_logsoftmax_kernel(
    const float* __restrict__ h,          // (n_nodes, 128)
    const float* __restrict__ W,          // (2, 256) row-major
    const float* __restrict__ bias,       // (2,)
    const int*   __restrict__ edge_index, // (n_edges, 2)
    float*       __restrict__ out,        // (n_edges, 2)
    int n_edges, int n_tiles)
{
    // 2 real W rows (512 floats) + 256-float zero bank for unused B columns.
    __shared__ float Wlds[512 + 256];

    const int t = threadIdx.x;

    // Stage W into LDS (each of 256 threads moves one float2), zero the pad bank.
    {
        v2f wv = *(const v2f*)(W + 2 * t);
        *(v2f*)(&Wlds[2 * t]) = wv;
        if (t < 128) {
            v2f z = {0.0f, 0.0f};
            *(v2f*)(&Wlds[512 + 2 * t]) = z;
        }
    }
    __syncthreads();

    const int wave = t >> 5;
    const int lane = t & 31;
    const int tile = blockIdx.x * WAVES_PER_BLOCK + wave;
    if (tile >= n_tiles) return;   // uniform per wave -> EXEC stays all-ones

    const int n  = lane & 15;   // B/C/D column index (classes live in n=0,1)
    const int hi = lane >> 4;   // half-wave select
    const int ko = hi * 2;      // K offset within each K=4 chunk (A/B layout)

    // Gather the edge endpoints for this lane's A-matrix row (clamped so
    // EXEC can stay full even on a ragged tail; reference size has no tail).
    int e  = tile * EDGES_PER_TILE + n;
    int ec = (e < n_edges) ? e : (n_edges - 1);
    const int u = edge_index[2 * ec + 0];
    const int v = edge_index[2 * ec + 1];
    const float* __restrict__ hu = h + (long long)u * D;
    const float* __restrict__ hv = h + (long long)v * D;

    // B source row in LDS: real W row for n<2, zero bank otherwise.
    const float* wrow = &Wlds[(n < 2) ? (n * 256) : 512];

    // Fold bias into the accumulator (column n has constant bias).
    const float bn = (n < 2) ? bias[n] : 0.0f;
    v8f acc = {bn, bn, bn, bn, bn, bn, bn, bn};

    // K = 0..127 : h[u] features
#pragma unroll
    for (int k = 0; k < 32; ++k) {
        v2f a  = *(const v2f*)(hu + 4 * k + ko);           // global_load_b64 gather
        v2f bw = *(const v2f*)(wrow + 4 * k + ko);         // ds_load_b64
        acc = __builtin_amdgcn_wmma_f32_16x16x4_f32(
            /*neg_a=*/false, a, /*neg_b=*/false, bw,
            /*c_mod=*/(short)0, acc, /*reuse_a=*/false, /*reuse_b=*/false);
    }
    // K = 128..255 : h[v] features
#pragma unroll
    for (int k = 0; k < 32; ++k) {
        v2f a  = *(const v2f*)(hv + 4 * k + ko);
        v2f bw = *(const v2f*)(wrow + D + 4 * k + ko);
        acc = __builtin_amdgcn_wmma_f32_16x16x4_f32(
            false, a, false, bw, (short)0, acc, false, false);
    }

    // D layout: VGPR r, lanes 0-15 -> (M=r, N=lane); lanes 16-31 -> (M=r+8).
    // Classes are columns 0 and 1 -> lane pairs (0,1) and (16,17) hold the
    // two logits of the same edge; exchange with an xor-1 shuffle.
    float res[8];
#pragma unroll
    for (int r = 0; r < 8; ++r) {
        float ls = acc[r];
        float lo = __shfl_xor(ls, 1, 32);
        float m  = fmaxf(ls, lo);
        res[r] = ls - m - logf(expf(ls - m) + expf(lo - m));
    }

    if (n < 2) {
        const int mbase = tile * EDGES_PER_TILE + hi * 8;
#pragma unroll
        for (int r = 0; r < 8; ++r) {
            int ee = mbase + r;
            if (ee < n_edges) out[2 * ee + n] = res[r];
        }
    }
}

extern "C" void kernel_launch(void* const* d_in, const int* in_sizes, int n_in,
                              void* d_out, int out_size, void* d_ws, size_t ws_size,
                              hipStream_t stream) {
    const float* h    = (const float*)d_in[0];
    const float* W    = (const float*)d_in[1];
    const float* b    = (const float*)d_in[2];
    const int*   ei   = (const int*)d_in[3];   // edge_index (n_edges, 2), int
    float*       out  = (float*)d_out;

    const int n_edges = in_sizes[3] / 2;
    const int n_tiles = (n_edges + EDGES_PER_TILE - 1) / EDGES_PER_TILE;
    const int blocks  = (n_tiles + WAVES_PER_BLOCK - 1) / WAVES_PER_BLOCK;

    mlp_edge_logsoftmax_kernel<<<blocks, 256, 0, stream>>>(
        h, W, b, ei, out, n_edges, n_tiles);
}